// RandomSquareDropout_30983894073613
// MI455X (gfx1250) — compile-verified
//
#include <hip/hip_runtime.h>

// Random-square dropout mask over [B=256, C=3, H=224, W=224] fp32.
// Pure streaming op: read 147MB + write 147MB, HBM-bound (~12.6us @ 23.3TB/s).
// Round 2: batch on blockIdx.y -> y_idx/x_idx become wave-uniform scalar
// loads (s_load, KMcnt) and the per-lane /37632 chain vanishes. Block tile
// (256 thr x 3 float4) divides the per-image float4 count 37632 exactly.

typedef float v4f __attribute__((ext_vector_type(4)));

namespace {
constexpr int SQ  = 32;
constexpr int Cn  = 3;
constexpr int Hn  = 224;
constexpr int Wn  = 224;
constexpr int W4  = Wn / 4;            // 56 float4 per row
constexpr int PER_B4 = Cn * Hn * W4;   // 37632 float4 per batch image
constexpr int THREADS = 256;           // 8 wave32
constexpr int VPT = 3;                 // float4 chunks per thread
constexpr int TILE = THREADS * VPT;    // 768; 37632 / 768 = 49 exact
}  // namespace

__global__ __launch_bounds__(THREADS) void RandomSquareDropout_kernel(
    const v4f* __restrict__ img,
    const int* __restrict__ y_idx,
    const int* __restrict__ x_idx,
    v4f* __restrict__ out) {
  const int b = blockIdx.y;            // batch image: uniform across block

  // Uniform -> scalar loads, resolved once per block via s_wait_kmcnt.
  const int yi = y_idx[b];
  const int xi = x_idx[b];

  const int base_in_img = blockIdx.x * TILE + threadIdx.x;
  const v4f* __restrict__ src = img + (size_t)b * PER_B4;
  v4f* __restrict__ dst       = out + (size_t)b * PER_B4;

#pragma unroll
  for (int k = 0; k < VPT; ++k) {
    int i = base_in_img + k * THREADS;     // float4 index within this image
    if (i >= PER_B4) break;                // never taken with exact tiling

    // Streaming load: data touched once, keep it out of the 192MB L2 (TH=NT).
    v4f v = __builtin_nontemporal_load(src + i);

    // Row/col within image: only small-constant div/mod remain per lane.
    int row = i / W4;                      // 0..671 (channel-stacked rows)
    int h   = row % Hn;                    // 0..223
    int w0  = (i - row * W4) * 4;          // column of component .x

    // Unsigned range trick: (h - yi) in [0,32) iff inside square rows.
    bool in_row = (unsigned)(h - yi) < (unsigned)SQ;

    // Per-component column test -> v_cndmask selects, no divergence.
    v.x = (in_row && (unsigned)(w0 + 0 - xi) < (unsigned)SQ) ? 0.0f : v.x;
    v.y = (in_row && (unsigned)(w0 + 1 - xi) < (unsigned)SQ) ? 0.0f : v.y;
    v.z = (in_row && (unsigned)(w0 + 2 - xi) < (unsigned)SQ) ? 0.0f : v.z;
    v.w = (in_row && (unsigned)(w0 + 3 - xi) < (unsigned)SQ) ? 0.0f : v.w;

    // Streaming store, non-temporal on the far cache as well.
    __builtin_nontemporal_store(v, dst + i);
  }
}

extern "C" void kernel_launch(void* const* d_in, const int* in_sizes, int n_in,
                              void* d_out, int out_size, void* d_ws, size_t ws_size,
                              hipStream_t stream) {
  (void)in_sizes; (void)n_in; (void)d_ws; (void)ws_size;

  const v4f* img  = (const v4f*)d_in[0];   // float32 [256,3,224,224]
  const int* yidx = (const int*)d_in[1];   // int32 [256]
  const int* xidx = (const int*)d_in[2];   // int32 [256]
  v4f* out        = (v4f*)d_out;           // float32, same shape as img

  const int n4      = out_size / 4;        // total float4 elements
  const int batches = n4 / PER_B4;         // 256
  const int xblocks = (PER_B4 + TILE - 1) / TILE;  // 49, exact

  dim3 grid(xblocks, batches, 1);
  RandomSquareDropout_kernel<<<grid, THREADS, 0, stream>>>(img, yidx, xidx, out);
}